// GNN_84189948936624
// MI455X (gfx1250) — compile-verified
//
#include <hip/hip_runtime.h>

typedef float v2f __attribute__((ext_vector_type(2)));
typedef float v8f __attribute__((ext_vector_type(8)));

#define D 128
#define MT 4  // 16-row tiles per wave in the GEMM (M=64 per block)

// ---------------------------------------------------------------------------
// Zero a float range (grid-stride).
// ---------------------------------------------------------------------------
__global__ void sage_zero_kernel(float* __restrict__ p, long long n) {
  long long i = (long long)blockIdx.x * blockDim.x + threadIdx.x;
  const long long stride = (long long)gridDim.x * blockDim.x;
  for (; i < n; i += stride) p[i] = 0.0f;
}

// ---------------------------------------------------------------------------
// Edge scatter: one wave32 per edge; each lane owns 4 contiguous columns.
// Gathers feat[src] (float4 -> GLOBAL_LOAD_B128) and atomically accumulates
// into accum[dst] (GLOBAL_ATOMIC_ADD_F32, RMW executes at L2 — the 25.6 MB
// feature matrix is L2-resident). Optionally accumulates degree (lane 0).
// ---------------------------------------------------------------------------
__global__ void sage_scatter_kernel(const float* __restrict__ feat,
                                    const long long* __restrict__ src,
                                    const long long* __restrict__ dst,
                                    float* __restrict__ accum,
                                    float* __restrict__ cnt,
                                    int n_edges, int add_count) {
  const int lane = threadIdx.x & 31;
  long long wave = ((long long)blockIdx.x * blockDim.x + threadIdx.x) >> 5;
  const long long nwaves = ((long long)gridDim.x * blockDim.x) >> 5;
  for (long long e = wave; e < n_edges; e += nwaves) {
    const long long s = src[e];
    const long long d = dst[e];
    const float4 v = ((const float4*)(feat + s * D))[lane];
    float* o = accum + d * D + (lane << 2);
    atomicAdd(o + 0, v.x);
    atomicAdd(o + 1, v.y);
    atomicAdd(o + 2, v.z);
    atomicAdd(o + 3, v.w);
    if (add_count && lane == 0) atomicAdd(cnt + d, 1.0f);
  }
}

// ---------------------------------------------------------------------------
// Fused SAGEConv GEMM on the f32 WMMA pipe:
//   out[r, :] = act( (S[r,:] / max(cnt[r],1)) @ Wl + bias + X[r,:] @ Wr )
//
// One wave computes MT=4 stacked 16x16 f32 tiles (M=64, N=16) with
// V_WMMA_F32_16X16X4_F32; each B fragment (weights, depends only on k,col)
// is loaded once and reused across the 4 row tiles -> 1 B-load + 4 A-loads
// per 4 WMMAs. Block = 256 threads = 8 waves covering 64 rows x 128 cols.
// Mean-normalization fused into A loads; bias + ReLU fused into writeback.
//
// f32 WMMA VGPR layouts (ISA 7.12.2):
//   A 16x4 : lanes 0-15 -> M=lane, {v0,v1}={K0,K1}; lanes 16-31 -> {K2,K3}
//   B  4x16: lanes 0-15 -> N=lane, {v0,v1}={K0,K1}; lanes 16-31 -> {K2,K3}
//   C 16x16: v_i: lanes 0-15 -> M=i, N=lane; lanes 16-31 -> M=i+8, N=lane-16
// Row indices are clamped (not predicated) so EXEC stays all-1s through the
// WMMAs on the ragged tail; only the final stores are guarded.
// ---------------------------------------------------------------------------
__global__ __launch_bounds__(256)
void sage_gemm_kernel(const float* __restrict__ S,    // neighbor sums [N, D]
                      const float* __restrict__ cnt,  // degrees [N]
                      const float* __restrict__ X,    // root features [N, D]
                      const float* __restrict__ Wl,   // [D, D] row-major
                      const float* __restrict__ Wr,   // [D, D] row-major
                      const float* __restrict__ bias, // [D]
                      float* __restrict__ out,        // [N, D]
                      int n_nodes, int do_relu) {
  const int lane = threadIdx.x & 31;
  const int waveId = threadIdx.x >> 5;
  const int rowBase = blockIdx.x * (16 * MT);
  const int colBase = waveId << 4;
  const int half = lane >> 4;  // selects K-pair for A/B, M-half for C
  const int lr = lane & 15;

  long long arow[MT];
  float inv[MT];
#pragma unroll
  for (int t = 0; t < MT; ++t) {
    int r = rowBase + 16 * t + lr;
    if (r > n_nodes - 1) r = n_nodes - 1;  // clamp: EXEC stays all-1s
    arow[t] = (long long)r * D;
    inv[t] = 1.0f / fmaxf(cnt[r], 1.0f);
  }

  v8f acc[MT];
#pragma unroll
  for (int t = 0; t < MT; ++t) acc[t] = (v8f){0.f,0.f,0.f,0.f,0.f,0.f,0.f,0.f};

  const float* __restrict__ WlCol = Wl + colBase + lr;
  const float* __restrict__ WrCol = Wr + colBase + lr;

  // Pass 1: (S / deg) @ Wl  — B fragment loaded once per k-step, reused x4.
#pragma unroll 4
  for (int k = 0; k < D; k += 4) {
    const int kk = k + half * 2;
    v2f b;
    b.x = WlCol[kk * D];
    b.y = WlCol[(kk + 1) * D];
#pragma unroll
    for (int t = 0; t < MT; ++t) {
      v2f a;
      a.x = S[arow[t] + kk] * inv[t];
      a.y = S[arow[t] + kk + 1] * inv[t];
      acc[t] = __builtin_amdgcn_wmma_f32_16x16x4_f32(false, a, false, b,
                                                     (short)0, acc[t],
                                                     false, false);
    }
  }
  // Pass 2: X @ Wr
#pragma unroll 4
  for (int k = 0; k < D; k += 4) {
    const int kk = k + half * 2;
    v2f b;
    b.x = WrCol[kk * D];
    b.y = WrCol[(kk + 1) * D];
#pragma unroll
    for (int t = 0; t < MT; ++t) {
      v2f a;
      a.x = X[arow[t] + kk];
      a.y = X[arow[t] + kk + 1];
      acc[t] = __builtin_amdgcn_wmma_f32_16x16x4_f32(false, a, false, b,
                                                     (short)0, acc[t],
                                                     false, false);
    }
  }

  const int col = colBase + lr;
  const float bv = bias[col];
#pragma unroll
  for (int t = 0; t < MT; ++t) {
#pragma unroll
    for (int i = 0; i < 8; ++i) {
      const int r = rowBase + 16 * t + i + half * 8;
      float v = acc[t][i] + bv;
      if (do_relu) v = fmaxf(v, 0.0f);
      if (r < n_nodes) out[(long long)r * D + col] = v;
    }
  }
}

// ---------------------------------------------------------------------------
// Orchestration:
//   zero(cnt, aggr) -> scatter(x)+count -> gemm1(relu) -> h
//   zero(aggr)      -> scatter(h)       -> gemm2       -> out
// ---------------------------------------------------------------------------
extern "C" void kernel_launch(void* const* d_in, const int* in_sizes, int n_in,
                              void* d_out, int out_size, void* d_ws, size_t ws_size,
                              hipStream_t stream) {
  (void)n_in; (void)out_size; (void)ws_size;

  const float* x      = (const float*)d_in[0];
  const long long* ei = (const long long*)d_in[1];  // int64 [2, E]
  const float* W1l    = (const float*)d_in[2];
  const float* b1     = (const float*)d_in[3];
  const float* W1r    = (const float*)d_in[4];
  const float* W2l    = (const float*)d_in[5];
  const float* b2     = (const float*)d_in[6];
  const float* W2r    = (const float*)d_in[7];
  float* out          = (float*)d_out;

  const int E = in_sizes[1] / 2;   // 800000
  const int N = in_sizes[0] / D;   // 50000
  const long long ND = (long long)N * D;

  float* ws   = (float*)d_ws;
  float* cnt  = ws;                                    // N floats (degrees)
  float* aggr = ws + (((long long)N + 127) & ~127LL);  // N*D floats (sums)
  float* h    = aggr + ND;                             // N*D (layer-1 out)

  const long long zeroLen = (long long)(aggr - ws) + ND;

  const dim3 blk(256);
  const int gemmGrid = (N + 16 * MT - 1) / (16 * MT);

  // Layer 1
  sage_zero_kernel<<<2048, blk, 0, stream>>>(ws, zeroLen);
  sage_scatter_kernel<<<8192, blk, 0, stream>>>(x, ei, ei + E, aggr, cnt, E, 1);
  sage_gemm_kernel<<<gemmGrid, blk, 0, stream>>>(aggr, cnt, x, W1l, W1r, b1, h,
                                                 N, 1);
  // Layer 2 (degrees already in cnt)
  sage_zero_kernel<<<2048, blk, 0, stream>>>(aggr, ND);
  sage_scatter_kernel<<<8192, blk, 0, stream>>>(h, ei, ei + E, aggr, cnt, E, 0);
  sage_gemm_kernel<<<gemmGrid, blk, 0, stream>>>(aggr, cnt, h, W2l, W2r, b2, out,
                                                 N, 0);
}